// MaskedWALoss_Guage_66640712564871
// MI455X (gfx1250) — compile-verified
//
#include <hip/hip_runtime.h>
#include <math.h>

#define ORB 54
#define F_P_C 0.0002f
#define F_Q_C 0.0001f
#define F_O_C 0.00015f

typedef float v2f __attribute__((ext_vector_type(2)));
typedef float v8f __attribute__((ext_vector_type(8)));

// ---------------------------------------------------------------------------
// Phase factors: R_tot = rint(inv(L^T) @ (edge_vec - (pos_dst - pos_src)))
// phase = exp(2*pi*i * R_tot . kpt)
// ---------------------------------------------------------------------------
__global__ void phase_kernel(const float* __restrict__ edge_vec,
                             const float* __restrict__ L,
                             const float* __restrict__ pos,
                             const int* __restrict__ src,
                             const int* __restrict__ dst,
                             const float* __restrict__ kpt,
                             float* __restrict__ phRe, float* __restrict__ phIm,
                             int E) {
  int e = blockIdx.x * blockDim.x + threadIdx.x;
  if (e >= E) return;
  // A = L^T (row-major), A[i][j] = L[j][i]
  float A00 = L[0], A01 = L[3], A02 = L[6];
  float A10 = L[1], A11 = L[4], A12 = L[7];
  float A20 = L[2], A21 = L[5], A22 = L[8];
  float c00 = A11 * A22 - A12 * A21;
  float c01 = A02 * A21 - A01 * A22;
  float c02 = A01 * A12 - A02 * A11;
  float c10 = A12 * A20 - A10 * A22;
  float c11 = A00 * A22 - A02 * A20;
  float c12 = A02 * A10 - A00 * A12;
  float c20 = A10 * A21 - A11 * A20;
  float c21 = A01 * A20 - A00 * A21;
  float c22 = A00 * A11 - A01 * A10;
  float det = A00 * c00 + A01 * c10 + A02 * c20;
  float inv = 1.0f / det;

  int s = src[e], d = dst[e];
  float r0 = edge_vec[e * 3 + 0] - (pos[d * 3 + 0] - pos[s * 3 + 0]);
  float r1 = edge_vec[e * 3 + 1] - (pos[d * 3 + 1] - pos[s * 3 + 1]);
  float r2 = edge_vec[e * 3 + 2] - (pos[d * 3 + 2] - pos[s * 3 + 2]);
  float t0 = (c00 * r0 + c01 * r1 + c02 * r2) * inv;
  float t1 = (c10 * r0 + c11 * r1 + c12 * r2) * inv;
  float t2 = (c20 * r0 + c21 * r1 + c22 * r2) * inv;
  float R0 = rintf(t0), R1 = rintf(t1), R2 = rintf(t2);
  float theta = 6.283185307179586f * (R0 * kpt[0] + R1 * kpt[1] + R2 * kpt[2]);
  float sn, cs;
  sincosf(theta, &sn, &cs);
  phRe[e] = cs;
  phIm[e] = sn;
}

// ---------------------------------------------------------------------------
// Scatter TRANSPOSED: Ht[dst*54+j][src*54+i] += phase_e * hr[e][i][j]
// (i.e. Ht = hk^T, so every GEMM reads B as [N x K] row-major).
// Edge block staged in LDS for coalesced global reads; atomics contiguous.
// ---------------------------------------------------------------------------
__global__ void scatter_hk(const float* __restrict__ hr,
                           const float* __restrict__ phRe,
                           const float* __restrict__ phIm,
                           const int* __restrict__ src,
                           const int* __restrict__ dst,
                           float* __restrict__ HtRe, float* __restrict__ HtIm,
                           int Mdim) {
  __shared__ float tile[ORB * ORB];
  int e = blockIdx.x;
  const float* blk = hr + (size_t)e * ORB * ORB;
  for (int idx = threadIdx.x; idx < ORB * ORB; idx += blockDim.x)
    tile[idx] = blk[idx];
  __syncthreads();
  float pr = phRe[e], pi = phIm[e];
  int s = src[e], d = dst[e];
  float* baseRe = HtRe + (size_t)(d * ORB) * Mdim + (size_t)(s * ORB);
  float* baseIm = HtIm + (size_t)(d * ORB) * Mdim + (size_t)(s * ORB);
  for (int idx = threadIdx.x; idx < ORB * ORB; idx += blockDim.x) {
    int j = idx / ORB;           // slow: column of hr block -> row of Ht
    int i = idx - j * ORB;       // fast: contiguous atomic addresses
    float v = tile[i * ORB + j];
    size_t off = (size_t)j * Mdim + i;
    atomicAdd(&baseRe[off], pr * v);
    atomicAdd(&baseIm[off], pi * v);
  }
}

// ---------------------------------------------------------------------------
// Complex GEMM, fp32 WMMA (v_wmma_f32_16x16x4_f32).
// C[M x N] = A @ B^T (complex), A:[M x K] row-major (conj if CONJ_A),
// B: [N x K] row-major.  Block tile 64x64, K-chunk 16, 256 thr = 8 waves,
// wave -> 16x32 of C.  float4 staging + register-prefetch pipeline.
// ---------------------------------------------------------------------------
template <bool CONJ_A>
__global__ __launch_bounds__(256) void cgemm64(
    const float* __restrict__ Are, const float* __restrict__ Aim,
    const float* __restrict__ Bre, const float* __restrict__ Bim,
    float* __restrict__ Cre, float* __restrict__ Cim, int Ndim, int Kdim) {
  __shared__ __align__(16) float sAre[64][20];
  __shared__ __align__(16) float sAim[64][20];
  __shared__ __align__(16) float sBre[64][20];
  __shared__ __align__(16) float sBim[64][20];

  const int tid = threadIdx.x;
  const int lane = tid & 31;
  const int wave = tid >> 5;
  const int wm = (wave & 3) * 16;   // wave row offset in 64-tile
  const int wn = (wave >> 2) * 32;  // wave col offset in 64-tile
  const int l15 = lane & 15;
  const int lh = lane >> 4;  // 0 or 1

  const int rBase = blockIdx.y * 64;
  const int nBase = blockIdx.x * 64;

  // Staging assignment: thread -> (row sr, 4 consecutive k at sk)
  const int sr = tid >> 2;        // 0..63
  const int sk = (tid & 3) * 4;   // 0,4,8,12
  const float* gAre = Are + (size_t)(rBase + sr) * Kdim + sk;
  const float* gAim = Aim + (size_t)(rBase + sr) * Kdim + sk;
  const float* gBre = Bre + (size_t)(nBase + sr) * Kdim + sk;
  const float* gBim = Bim + (size_t)(nBase + sr) * Kdim + sk;

  float4 pAre = *(const float4*)gAre;
  float4 pAim = *(const float4*)gAim;
  float4 pBre = *(const float4*)gBre;
  float4 pBim = *(const float4*)gBim;

  v8f accRe0 = {}, accRe1 = {}, accIm0 = {}, accIm1 = {};

  for (int kb = 0; kb < Kdim; kb += 16) {
    __syncthreads();  // protect LDS from previous chunk's readers
    *(float4*)&sAre[sr][sk] = pAre;
    float4 t = pAim;
    if (CONJ_A) { t.x = -t.x; t.y = -t.y; t.z = -t.z; t.w = -t.w; }
    *(float4*)&sAim[sr][sk] = t;
    *(float4*)&sBre[sr][sk] = pBre;
    *(float4*)&sBim[sr][sk] = pBim;
    __syncthreads();

    // Prefetch next chunk while WMMAs run on this one.
    int kn = kb + 16;
    if (kn < Kdim) {
      pAre = *(const float4*)(gAre + kn);
      pAim = *(const float4*)(gAim + kn);
      pBre = *(const float4*)(gBre + kn);
      pBim = *(const float4*)(gBim + kn);
    }

#pragma unroll
    for (int kk = 0; kk < 16; kk += 4) {
      const int kh = kk + 2 * lh;
      const int mrow = wm + l15;
      v2f aRe, aIm, aImN;
      aRe.x = sAre[mrow][kh];
      aRe.y = sAre[mrow][kh + 1];
      aIm.x = sAim[mrow][kh];
      aIm.y = sAim[mrow][kh + 1];
      aImN = -aIm;
      {
        int nc = wn + l15;
        v2f bRe, bIm;
        bRe.x = sBre[nc][kh];
        bRe.y = sBre[nc][kh + 1];
        bIm.x = sBim[nc][kh];
        bIm.y = sBim[nc][kh + 1];
        accRe0 = __builtin_amdgcn_wmma_f32_16x16x4_f32(false, aRe, false, bRe,
                                                       (short)0, accRe0, false, false);
        accRe0 = __builtin_amdgcn_wmma_f32_16x16x4_f32(false, aImN, false, bIm,
                                                       (short)0, accRe0, false, false);
        accIm0 = __builtin_amdgcn_wmma_f32_16x16x4_f32(false, aRe, false, bIm,
                                                       (short)0, accIm0, false, false);
        accIm0 = __builtin_amdgcn_wmma_f32_16x16x4_f32(false, aIm, false, bRe,
                                                       (short)0, accIm0, false, false);
      }
      {
        int nc = wn + 16 + l15;
        v2f bRe, bIm;
        bRe.x = sBre[nc][kh];
        bRe.y = sBre[nc][kh + 1];
        bIm.x = sBim[nc][kh];
        bIm.y = sBim[nc][kh + 1];
        accRe1 = __builtin_amdgcn_wmma_f32_16x16x4_f32(false, aRe, false, bRe,
                                                       (short)0, accRe1, false, false);
        accRe1 = __builtin_amdgcn_wmma_f32_16x16x4_f32(false, aImN, false, bIm,
                                                       (short)0, accRe1, false, false);
        accIm1 = __builtin_amdgcn_wmma_f32_16x16x4_f32(false, aRe, false, bIm,
                                                       (short)0, accIm1, false, false);
        accIm1 = __builtin_amdgcn_wmma_f32_16x16x4_f32(false, aIm, false, bRe,
                                                       (short)0, accIm1, false, false);
      }
    }
  }

  // Epilogue: C/D layout — VGPR r: lanes 0-15 -> M=r, lanes 16-31 -> M=r+8.
#pragma unroll
  for (int r = 0; r < 8; ++r) {
    int row = rBase + wm + r + 8 * lh;
    int col = nBase + wn + l15;
    size_t g0 = (size_t)row * Ndim + col;
    Cre[g0] = accRe0[r];
    Cim[g0] = accIm0[r];
    Cre[g0 + 16] = accRe1[r];
    Cim[g0 + 16] = accIm1[r];
  }
}

// ---------------------------------------------------------------------------
// Reductions (deterministic two-stage).
// ---------------------------------------------------------------------------
__device__ inline float block_reduce(float v, float* sbuf) {
  int t = threadIdx.x;
  sbuf[t] = v;
  __syncthreads();
  for (int s = blockDim.x >> 1; s > 0; s >>= 1) {
    if (t < s) sbuf[t] += sbuf[t + s];
    __syncthreads();
  }
  float r = sbuf[0];
  __syncthreads();
  return r;
}

__global__ void reduce_pass1(const float* __restrict__ Hgt,
                             const float* __restrict__ Hpred,
                             const float* __restrict__ ovl,
                             const float* __restrict__ mask,
                             size_t n, float* __restrict__ part, int G) {
  __shared__ float sbuf[256];
  float s0 = 0.f, s1 = 0.f, s2 = 0.f;
  for (size_t i = (size_t)blockIdx.x * blockDim.x + threadIdx.x; i < n;
       i += (size_t)G * blockDim.x) {
    float o = ovl[i];
    s0 += (Hpred[i] - Hgt[i]) * o;
    s1 += o * o;
    s2 += mask[i];
  }
  s0 = block_reduce(s0, sbuf);
  s1 = block_reduce(s1, sbuf);
  s2 = block_reduce(s2, sbuf);
  if (threadIdx.x == 0) {
    part[blockIdx.x] = s0;
    part[G + blockIdx.x] = s1;
    part[2 * G + blockIdx.x] = s2;
  }
}

__global__ void mu_kernel(const float* __restrict__ part1, int G,
                          const float* __restrict__ rP1re,
                          const float* __restrict__ rP2re, int NP,
                          const float* __restrict__ rQ1re,
                          const float* __restrict__ rQ2re, int NQ,
                          float* __restrict__ scal) {
  __shared__ float sbuf[256];
  float s0 = 0.f, s1 = 0.f, s2 = 0.f, tp = 0.f, tq = 0.f;
  for (int i = threadIdx.x; i < G; i += blockDim.x) {
    s0 += part1[i];
    s1 += part1[G + i];
    s2 += part1[2 * G + i];
  }
  for (int i = threadIdx.x; i < NP; i += blockDim.x)
    tp += rP2re[(size_t)i * NP + i] - rP1re[(size_t)i * NP + i];
  for (int i = threadIdx.x; i < NQ; i += blockDim.x)
    tq += rQ2re[(size_t)i * NQ + i] - rQ1re[(size_t)i * NQ + i];
  s0 = block_reduce(s0, sbuf);
  s1 = block_reduce(s1, sbuf);
  s2 = block_reduce(s2, sbuf);
  tp = block_reduce(tp, sbuf);
  tq = block_reduce(tq, sbuf);
  if (threadIdx.x == 0) {
    float fR = 1.0f - F_P_C - F_Q_C - F_O_C;
    float Nm = s2;
    float n1 = fR * s0 / Nm;
    float d1 = fR * s1 / Nm;
    float n2 = F_P_C * tp / ((float)NP * (float)NP);
    float n3 = F_Q_C * tq / ((float)NQ * (float)NQ);
    float d2 = F_P_C / (float)NP;
    float d3 = F_Q_C / (float)NQ;
    scal[0] = Nm;
    scal[1] = (n1 + n2 + n3) / (d1 + d2 + d3);  // mu
  }
}

__global__ void loss_hr_kernel(const float* __restrict__ Hgt,
                               const float* __restrict__ Hpred,
                               const float* __restrict__ ovl,
                               const float* __restrict__ scal, size_t n,
                               float* __restrict__ part, int G) {
  __shared__ float sbuf[256];
  float mu = scal[1];
  float s = 0.f;
  for (size_t i = (size_t)blockIdx.x * blockDim.x + threadIdx.x; i < n;
       i += (size_t)G * blockDim.x)
    s += fabsf(Hgt[i] + mu * ovl[i] - Hpred[i]);
  s = block_reduce(s, sbuf);
  if (threadIdx.x == 0) part[blockIdx.x] = s;
}

__global__ void loss_mat_kernel(const float* __restrict__ r1re,
                                const float* __restrict__ r1im,
                                const float* __restrict__ r2re,
                                const float* __restrict__ r2im,
                                const float* __restrict__ scal, int rows,
                                int cols, int useEye, float* __restrict__ part,
                                int G) {
  __shared__ float sbuf[256];
  float mu = useEye ? scal[1] : 0.0f;
  size_t n = (size_t)rows * cols;
  float s = 0.f;
  for (size_t i = (size_t)blockIdx.x * blockDim.x + threadIdx.x; i < n;
       i += (size_t)G * blockDim.x) {
    int r = (int)(i / cols);
    int c = (int)(i - (size_t)r * cols);
    float eye = (useEye && r == c) ? mu : 0.0f;
    s += fabsf(r1re[i] + eye - r2re[i]) + fabsf(r1im[i] - r2im[i]);
  }
  s = block_reduce(s, sbuf);
  if (threadIdx.x == 0) part[blockIdx.x] = s;
}

__global__ void final_kernel(const float* __restrict__ partHR, int GHR,
                             const float* __restrict__ partP,
                             const float* __restrict__ partQ,
                             const float* __restrict__ partPQ, int GM,
                             const float* __restrict__ scal, int NP, int NQ,
                             float* __restrict__ out) {
  __shared__ float sbuf[256];
  float a = 0.f, b = 0.f, c = 0.f, d = 0.f;
  for (int i = threadIdx.x; i < GHR; i += blockDim.x) a += partHR[i];
  for (int i = threadIdx.x; i < GM; i += blockDim.x) {
    b += partP[i];
    c += partQ[i];
    d += partPQ[i];
  }
  a = block_reduce(a, sbuf);
  b = block_reduce(b, sbuf);
  c = block_reduce(c, sbuf);
  d = block_reduce(d, sbuf);
  if (threadIdx.x == 0) {
    float fR = 1.0f - F_P_C - F_Q_C - F_O_C;
    float Nm = scal[0];
    out[0] = fR * a / Nm + F_P_C * b / ((float)NP * (float)NP) +
             F_Q_C * c / ((float)NQ * (float)NQ) +
             F_O_C * d / ((float)NP * (float)NQ);
  }
}

// ---------------------------------------------------------------------------
extern "C" void kernel_launch(void* const* d_in, const int* in_sizes, int n_in,
                              void* d_out, int out_size, void* d_ws,
                              size_t ws_size, hipStream_t stream) {
  const float* edge_vec = (const float*)d_in[0];
  const float* lattice = (const float*)d_in[1];
  const float* pos = (const float*)d_in[2];
  const float* hr_gt = (const float*)d_in[3];
  const float* hr_pred = (const float*)d_in[4];
  const float* kpt = (const float*)d_in[5];
  const float* P_re = (const float*)d_in[6];
  const float* P_im = (const float*)d_in[7];
  const float* Q_re = (const float*)d_in[8];
  const float* Q_im = (const float*)d_in[9];
  const float* H_gt = (const float*)d_in[10];
  const float* pred_H = (const float*)d_in[11];
  const float* overlap = (const float*)d_in[12];
  const float* mask = (const float*)d_in[13];
  const int* esrc = (const int*)d_in[14];
  const int* edst = (const int*)d_in[15];

  const int E = in_sizes[14];
  const int nAtoms = in_sizes[2] / 3;
  const int M = ORB * nAtoms;      // 3456
  const int NP = in_sizes[6] / M;  // 256
  const int NQ = in_sizes[8] / M;  // 384

  float* ws = (float*)d_ws;
  size_t off = 0;
  auto take = [&](size_t n) {
    float* p = ws + off;
    off += n;
    return p;
  };
  float* phRe = take(E);
  float* phIm = take(E);
  float* HtRe = take((size_t)M * M);
  float* HtIm = take((size_t)M * M);
  float* WpRe = take((size_t)NP * M);
  float* WpIm = take((size_t)NP * M);
  float* WqRe = take((size_t)NQ * M);
  float* WqIm = take((size_t)NQ * M);
  float* rPre[2] = {take((size_t)NP * NP), take((size_t)NP * NP)};
  float* rPim[2] = {take((size_t)NP * NP), take((size_t)NP * NP)};
  float* rQre[2] = {take((size_t)NQ * NQ), take((size_t)NQ * NQ)};
  float* rQim[2] = {take((size_t)NQ * NQ), take((size_t)NQ * NQ)};
  float* rPQre[2] = {take((size_t)NP * NQ), take((size_t)NP * NQ)};
  float* rPQim[2] = {take((size_t)NP * NQ), take((size_t)NP * NQ)};
  const int G1 = 1024, GM = 64;
  float* part1 = take(3 * (size_t)G1);
  float* partHR = take(G1);
  float* partP = take(GM);
  float* partQ = take(GM);
  float* partPQ = take(GM);
  float* scal = take(8);
  if (off * sizeof(float) > ws_size) return;  // insufficient scratch

  phase_kernel<<<(E + 255) / 256, 256, 0, stream>>>(
      edge_vec, lattice, pos, esrc, edst, kpt, phRe, phIm, E);

  const size_t nHR = (size_t)E * ORB * ORB;
  const float* hrs[2] = {hr_gt, hr_pred};

  for (int p = 0; p < 2; ++p) {
    hipMemsetAsync(HtRe, 0, (size_t)M * M * sizeof(float), stream);
    hipMemsetAsync(HtIm, 0, (size_t)M * M * sizeof(float), stream);
    scatter_hk<<<E, 256, 0, stream>>>(hrs[p], phRe, phIm, esrc, edst, HtRe,
                                      HtIm, M);
    // Stage 1: W = conj(P) @ hk = conj(P) @ Ht^T   (B = Ht, [N=M x K=M])
    cgemm64<true><<<dim3(M / 64, NP / 64), 256, 0, stream>>>(
        P_re, P_im, HtRe, HtIm, WpRe, WpIm, M, M);
    cgemm64<true><<<dim3(M / 64, NQ / 64), 256, 0, stream>>>(
        Q_re, Q_im, HtRe, HtIm, WqRe, WqIm, M, M);
    // Stage 2: rP = Wp @ P^T, rQ = Wq @ Q^T, rPQ = Wp @ Q^T
    cgemm64<false><<<dim3(NP / 64, NP / 64), 256, 0, stream>>>(
        WpRe, WpIm, P_re, P_im, rPre[p], rPim[p], NP, M);
    cgemm64<false><<<dim3(NQ / 64, NQ / 64), 256, 0, stream>>>(
        WqRe, WqIm, Q_re, Q_im, rQre[p], rQim[p], NQ, M);
    cgemm64<false><<<dim3(NQ / 64, NP / 64), 256, 0, stream>>>(
        WpRe, WpIm, Q_re, Q_im, rPQre[p], rPQim[p], NQ, M);
  }

  reduce_pass1<<<G1, 256, 0, stream>>>(H_gt, pred_H, overlap, mask, nHR, part1,
                                       G1);
  mu_kernel<<<1, 256, 0, stream>>>(part1, G1, rPre[0], rPre[1], NP, rQre[0],
                                   rQre[1], NQ, scal);
  loss_hr_kernel<<<G1, 256, 0, stream>>>(H_gt, pred_H, overlap, scal, nHR,
                                         partHR, G1);
  loss_mat_kernel<<<GM, 256, 0, stream>>>(rPre[0], rPim[0], rPre[1], rPim[1],
                                          scal, NP, NP, 1, partP, GM);
  loss_mat_kernel<<<GM, 256, 0, stream>>>(rQre[0], rQim[0], rQre[1], rQim[1],
                                          scal, NQ, NQ, 1, partQ, GM);
  loss_mat_kernel<<<GM, 256, 0, stream>>>(rPQre[0], rPQim[0], rPQre[1],
                                          rPQim[1], scal, NP, NQ, 0, partPQ,
                                          GM);
  final_kernel<<<1, 256, 0, stream>>>(partHR, G1, partP, partQ, partPQ, GM,
                                      scal, NP, NQ, (float*)d_out);
}